// SAGE_3255585210654
// MI455X (gfx1250) — compile-verified
//
#include <hip/hip_runtime.h>
#include <math.h>

typedef __attribute__((ext_vector_type(2))) float v2f;
typedef __attribute__((ext_vector_type(8))) float v8f;

#define NFEAT 128
#define HID   16
#define NCLS  10

// ---------------- utility ----------------
__global__ void zero_f32(float* __restrict__ p, long n) {
  long i = (long)blockIdx.x * blockDim.x + threadIdx.x;
  if (i < n) p[i] = 0.f;
}

// ---------------- pad layer-3 weights [10,16] -> [16,16], bias [10] -> [16] ----------------
__global__ void pad_w3(const float* __restrict__ W3l, const float* __restrict__ b3,
                       const float* __restrict__ W3r,
                       float* __restrict__ Wlp, float* __restrict__ bp,
                       float* __restrict__ Wrp) {
  int t = threadIdx.x;            // 256 threads
  int o = t >> 4;
  Wlp[t] = (o < NCLS) ? W3l[t] : 0.f;   // row-major [o][k]; rows 0-9 copied verbatim
  Wrp[t] = (o < NCLS) ? W3r[t] : 0.f;
  if (t < HID) bp[t] = (t < NCLS) ? b3[t] : 0.f;
}

// ---------------- degree ----------------
__global__ void deg_kernel(const long long* __restrict__ dst, long E,
                           float* __restrict__ deg) {
  long e = (long)blockIdx.x * blockDim.x + threadIdx.x;
  if (e < E) atomicAdd(&deg[dst[e]], 1.0f);
}

__global__ void invdeg_kernel(float* __restrict__ deg, int n) {
  int i = blockIdx.x * blockDim.x + threadIdx.x;
  if (i < n) deg[i] = 1.0f / fmaxf(deg[i], 1.0f);
}

// ---------------- argmax + embedding gather ----------------
// one wave32 per node; lane l covers features l, l+32, l+64, l+96
__global__ void argmax_embed(const float* __restrict__ x,
                             const float* __restrict__ emb,
                             float* __restrict__ h, int n_nodes) {
  int wid  = threadIdx.x >> 5;
  int lane = threadIdx.x & 31;
  int node = blockIdx.x * (blockDim.x >> 5) + wid;
  if (node >= n_nodes) return;
  const float* row = x + (long)node * NFEAT;
  float bv = -INFINITY; int bi = 0;
#pragma unroll
  for (int j = 0; j < 4; ++j) {
    int idx = lane + 32 * j;
    float v = row[idx];
    if (v > bv) { bv = v; bi = idx; }   // per-lane indices ascend -> first occurrence kept
  }
  // butterfly reduction; prefer smaller index on ties (jnp.argmax semantics)
#pragma unroll
  for (int m = 16; m; m >>= 1) {
    float ov = __shfl_xor(bv, m);
    int   oi = __shfl_xor(bi, m);
    if (ov > bv || (ov == bv && oi < bi)) { bv = ov; bi = oi; }
  }
  if (lane < HID) h[(long)node * HID + lane] = emb[bi * HID + lane];
}

// ---------------- edge scatter: agg[dst] += h[src] ----------------
// 16 lanes per edge (one feature each): contiguous 64B loads + contiguous atomics
__global__ void scatter_kernel(const long long* __restrict__ src,
                               const long long* __restrict__ dst,
                               const float* __restrict__ hin,
                               float* __restrict__ agg, long E) {
  long t = (long)blockIdx.x * blockDim.x + threadIdx.x;
  long e = t >> 4;
  int  f = (int)(t & 15);
  if (e < E) {
    long s = src[e];
    long d = dst[e];
    atomicAdd(&agg[d * HID + f], hin[s * HID + f]);
  }
}

// ---------------- SAGE transform via WMMA f32 16x16x4 ----------------
// one wave per 16-node tile: D = (agg*invdeg) @ Wl^T + bias + h @ Wr^T,
// optional ReLU, then per-row L2 normalization. Weights always dense 16x16.
__global__ void sage_transform(const float* __restrict__ hin,
                               const float* __restrict__ agg,
                               const float* __restrict__ invdeg,
                               const float* __restrict__ Wl,
                               const float* __restrict__ bias,
                               const float* __restrict__ Wr,
                               float* __restrict__ hout,
                               int n_nodes, int do_relu) {
  int wid  = threadIdx.x >> 5;
  int lane = threadIdx.x & 31;
  int tile = blockIdx.x * (blockDim.x >> 5) + wid;
  if (tile * 16 >= n_nodes) return;          // whole-wave exit keeps EXEC full for WMMA

  int row  = lane & 15;        // A-matrix M row held by this lane
  int half = lane >> 4;        // which K pair (A/B layout) / which M half (C/D layout)
  int kb   = half * 2;
  int o    = lane & 15;        // B/C/D column (output feature)
  long node = (long)tile * 16 + row;

  float idg = invdeg[node];
  float bb  = bias[o];
  v8f c = { bb, bb, bb, bb, bb, bb, bb, bb };

  const float* arow = agg + node * HID;
  const float* hrow = hin + node * HID;
  const float* wlrow = Wl + o * HID;
  const float* wrrow = Wr + o * HID;

  // acc += mean_agg @ Wl^T   (B[k][o] = Wl[o*16+k])
#pragma unroll
  for (int kc = 0; kc < 4; ++kc) {
    int k = kc * 4 + kb;
    v2f a = *(const v2f*)(arow + k);
    a.x *= idg; a.y *= idg;
    v2f b = *(const v2f*)(wlrow + k);
    c = __builtin_amdgcn_wmma_f32_16x16x4_f32(false, a, false, b, (short)0, c,
                                              false, false);
  }
  // acc += h @ Wr^T
#pragma unroll
  for (int kc = 0; kc < 4; ++kc) {
    int k = kc * 4 + kb;
    v2f a = *(const v2f*)(hrow + k);
    v2f b = *(const v2f*)(wrrow + k);
    c = __builtin_amdgcn_wmma_f32_16x16x4_f32(false, a, false, b, (short)0, c,
                                              false, false);
  }

  if (do_relu) {
#pragma unroll
    for (int v = 0; v < 8; ++v) c[v] = fmaxf(c[v], 0.f);
  }

  // per-row L2 norm: VGPR v holds row v (lanes 0-15) and row v+8 (lanes 16-31);
  // xor-shuffles with mask < 16 stay within each 16-lane group.
#pragma unroll
  for (int v = 0; v < 8; ++v) {
    float s = c[v] * c[v];
    s += __shfl_xor(s, 1);
    s += __shfl_xor(s, 2);
    s += __shfl_xor(s, 4);
    s += __shfl_xor(s, 8);
    float nrm = sqrtf(s);
    c[v] = c[v] / fmaxf(nrm, 1e-12f);
  }

  long base = ((long)tile * 16 + half * 8) * HID;
#pragma unroll
  for (int v = 0; v < 8; ++v) {
    hout[base + (long)v * HID + o] = c[v];
  }
}

// ---------------- global mean pool (atomics) ----------------
__global__ void pool_kernel(const float* __restrict__ h,
                            const long long* __restrict__ batch,
                            float* __restrict__ pooled,
                            float* __restrict__ cnt, int n_nodes) {
  long t = (long)blockIdx.x * blockDim.x + threadIdx.x;
  int n = (int)(t >> 4);
  int cidx = (int)(t & 15);
  if (n < n_nodes) {
    long g = batch[n];
    if (cidx < NCLS) atomicAdd(&pooled[g * NCLS + cidx], h[(long)n * HID + cidx]);
    if (cidx == 0)   atomicAdd(&cnt[g], 1.0f);
  }
}

// ---------------- mean + softmax ----------------
__global__ void softmax_kernel(const float* __restrict__ pooled,
                               const float* __restrict__ cnt,
                               float* __restrict__ out, int ngraphs) {
  int g = blockIdx.x * blockDim.x + threadIdx.x;
  if (g >= ngraphs) return;
  float inv = 1.0f / fmaxf(cnt[g], 1.0f);
  float vals[NCLS];
  float mx = -INFINITY;
#pragma unroll
  for (int c = 0; c < NCLS; ++c) {
    vals[c] = pooled[g * NCLS + c] * inv;
    mx = fmaxf(mx, vals[c]);
  }
  float s = 0.f;
#pragma unroll
  for (int c = 0; c < NCLS; ++c) { vals[c] = __expf(vals[c] - mx); s += vals[c]; }
  float is = 1.0f / s;
#pragma unroll
  for (int c = 0; c < NCLS; ++c) out[g * NCLS + c] = vals[c] * is;
}

// ---------------- launch ----------------
extern "C" void kernel_launch(void* const* d_in, const int* in_sizes, int n_in,
                              void* d_out, int out_size, void* d_ws, size_t ws_size,
                              hipStream_t stream) {
  const float*     x     = (const float*)d_in[0];
  const long long* ei    = (const long long*)d_in[1];
  const long long* batch = (const long long*)d_in[2];
  const float*     emb   = (const float*)d_in[3];
  const float* W1l = (const float*)d_in[4];
  const float* b1  = (const float*)d_in[5];
  const float* W1r = (const float*)d_in[6];
  const float* W2l = (const float*)d_in[7];
  const float* b2  = (const float*)d_in[8];
  const float* W2r = (const float*)d_in[9];
  const float* W3l = (const float*)d_in[10];
  const float* b3  = (const float*)d_in[11];
  const float* W3r = (const float*)d_in[12];

  int  n_nodes = in_sizes[0] / NFEAT;
  long E       = (long)in_sizes[1] / 2;
  const long long* src = ei;
  const long long* dst = ei + E;
  int ngraphs = out_size / NCLS;

  float* ws     = (float*)d_ws;
  float* hA     = ws;
  float* hB     = hA + (long)n_nodes * HID;
  float* agg    = hB + (long)n_nodes * HID;
  float* invdeg = agg + (long)n_nodes * HID;
  float* pooled = invdeg + n_nodes;
  float* cnt    = pooled + (long)ngraphs * NCLS;
  float* W3lp   = cnt + ngraphs;          // padded layer-3 params: 256 + 256 + 16 floats
  float* W3rp   = W3lp + HID * HID;
  float* b3p    = W3rp + HID * HID;
  float* out    = (float*)d_out;

  const int B = 256;

  // pad layer-3 weights to dense 16x16 (zeros in rows 10-15)
  pad_w3<<<1, 256, 0, stream>>>(W3l, b3, W3r, W3lp, b3p, W3rp);

  // degrees (shared across all three convs)
  zero_f32<<<(n_nodes + B - 1) / B, B, 0, stream>>>(invdeg, n_nodes);
  deg_kernel<<<(unsigned)((E + B - 1) / B), B, 0, stream>>>(dst, E, invdeg);
  invdeg_kernel<<<(n_nodes + B - 1) / B, B, 0, stream>>>(invdeg, n_nodes);

  // h0 = emb[argmax(x)]
  argmax_embed<<<(n_nodes + 7) / 8, B, 0, stream>>>(x, emb, hA, n_nodes);

  const float* Wls[3] = { W1l, W2l, W3lp };
  const float* bs[3]  = { b1,  b2,  b3p  };
  const float* Wrs[3] = { W1r, W2r, W3rp };

  float* hin = hA;
  float* hout = hB;
  int tiles = (n_nodes + 15) / 16;
  for (int L = 0; L < 3; ++L) {
    zero_f32<<<(unsigned)(((long)n_nodes * HID + B - 1) / B), B, 0, stream>>>(agg, (long)n_nodes * HID);
    scatter_kernel<<<(unsigned)((E * 16 + B - 1) / B), B, 0, stream>>>(src, dst, hin, agg, E);
    sage_transform<<<(tiles + 7) / 8, B, 0, stream>>>(hin, agg, invdeg,
                                                      Wls[L], bs[L], Wrs[L],
                                                      hout, n_nodes, (L < 2) ? 1 : 0);
    float* t = hin; hin = hout; hout = t;
  }

  // pooled + cnt are contiguous -> one zero pass
  zero_f32<<<((long)ngraphs * NCLS + ngraphs + B - 1) / B, B, 0, stream>>>(pooled, (long)ngraphs * NCLS + ngraphs);
  pool_kernel<<<(unsigned)(((long)n_nodes * 16 + B - 1) / B), B, 0, stream>>>(hin, batch, pooled, cnt, n_nodes);
  softmax_kernel<<<(ngraphs + B - 1) / B, B, 0, stream>>>(pooled, cnt, out, ngraphs);
}